// L2RU_34857954574985
// MI455X (gfx1250) — compile-verified
//
#include <hip/hip_runtime.h>

// ---------------------------------------------------------------------------
// L2RU on MI455X (gfx1250, wave32, WMMA).
// Dims: n=256, Bsz=8, S=8192.  All math fp32 via V_WMMA_F32_16X16X4_F32.
// ---------------------------------------------------------------------------

typedef __attribute__((ext_vector_type(2))) float v2f;
typedef __attribute__((ext_vector_type(8))) float v8f;

#define NDIM 256
#define BSZ 8
#define SEQ 8192
#define S1 (SEQ + 1)
#define LCHUNK 64
#define CCHUNK 128
#define SLOT 65536  // floats per 256x256 matrix slot

__device__ __forceinline__ v8f wmma4(v2f a, v2f b, v8f c) {
  // D(16x16,f32) = A(16x4,f32) * B(4x16,f32) + C
  return __builtin_amdgcn_wmma_f32_16x16x4_f32(
      /*neg_a=*/false, a, /*neg_b=*/false, b,
      /*c_mod=*/(short)0, c, /*reuse_a=*/false, /*reuse_b=*/false);
}

// ---------------------------------------------------------------------------
// Generic WMMA GEMM:  C[z] = op(A[z]) * op(B[z]) + beta*C[z]
// TA/TB are compile-time so the K-loop is branch-free; contiguous operand
// pairs load as b64. Two accumulators halve the dependent-WMMA chain.
// M,N multiples of 16; K multiple of 8.
// ---------------------------------------------------------------------------
template <int TA, int TB>
__global__ void gg_kernel(const float* __restrict__ A, long lda, long sA,
                          const float* __restrict__ B, long ldb, long sB,
                          float* __restrict__ C, long ldc, long sC,
                          int M, int N, int K, float beta) {
  const int lane = threadIdx.x & 31;
  const int wv = threadIdx.x >> 5;
  const int ntiles = N >> 4;
  const long tiles = (long)(M >> 4) * ntiles;
  long tile = (long)blockIdx.x * (blockDim.x >> 5) + wv;
  if (tile >= tiles) return;  // wave-uniform: EXEC stays all-1 for WMMA
  const int nt = (int)(tile % ntiles);
  const int mt = (int)(tile / ntiles);
  const float* Ab = A + (long)blockIdx.z * sA;
  const float* Bb = B + (long)blockIdx.z * sB;
  float* Cb = C + (long)blockIdx.z * sC;
  const int hi = lane >> 4, lo = lane & 15;
  const int am = mt * 16 + lo;  // A-operand row (M index) for this lane
  const int bn = nt * 16 + lo;  // B-operand col (N index) for this lane
  const float* aRow = Ab + (TA ? (long)am : (long)am * lda);  // row/col base
  const float* bRow = Bb + (TB ? (long)bn * ldb : (long)bn);
  v8f acc0 = {0.f, 0.f, 0.f, 0.f, 0.f, 0.f, 0.f, 0.f};
  v8f acc1 = {0.f, 0.f, 0.f, 0.f, 0.f, 0.f, 0.f, 0.f};
  for (int k = 0; k < K; k += 8) {
    const int ks0 = k + 2 * hi;
    const int ks1 = ks0 + 4;
    v2f a0, a1, b0, b1;
    if (TA) {  // strided scalar loads
      a0.x = aRow[(long)ks0 * lda];
      a0.y = aRow[(long)(ks0 + 1) * lda];
      a1.x = aRow[(long)ks1 * lda];
      a1.y = aRow[(long)(ks1 + 1) * lda];
    } else {  // contiguous pair -> b64
      a0 = *(const v2f*)(aRow + ks0);
      a1 = *(const v2f*)(aRow + ks1);
    }
    if (TB) {  // B^T: contiguous pair -> b64
      b0 = *(const v2f*)(bRow + ks0);
      b1 = *(const v2f*)(bRow + ks1);
    } else {
      b0.x = bRow[(long)ks0 * ldb];
      b0.y = bRow[(long)(ks0 + 1) * ldb];
      b1.x = bRow[(long)ks1 * ldb];
      b1.y = bRow[(long)(ks1 + 1) * ldb];
    }
    acc0 = wmma4(a0, b0, acc0);
    acc1 = wmma4(a1, b1, acc1);
  }
#pragma unroll
  for (int r = 0; r < 8; ++r) {
    const int m = mt * 16 + r + 8 * hi;  // D layout: VGPR r -> rows r, r+8
    const long idx = (long)m * ldc + bn;
    float o = acc0[r] + acc1[r];
    if (beta != 0.f) o += beta * Cb[idx];
    Cb[idx] = o;
  }
}

// ---------------------------------------------------------------------------
// Chunk-local scan: block c runs L steps of x <- x*A^T + v_t from zero state.
// B-operand fragment needs (A^T)[k][n] = A[n][k]: contiguous in k for
// row-major A, so the 128-VGPR fragment preload is 64 b64 loads per lane.
// State ping-pongs in LDS (rows 8..15 zero); A-fragments are ds_load_b64.
// Two interleaved accumulators halve the dependent-WMMA chain.
// ---------------------------------------------------------------------------
__global__ __launch_bounds__(512) void chunk_scan(const float* __restrict__ v,
                                                  const float* __restrict__ Amat,
                                                  float* __restrict__ states,
                                                  float* __restrict__ est) {
  const int lane = threadIdx.x & 31;
  const int w = threadIdx.x >> 5;  // 0..15: output column tile
  const int c = blockIdx.x;
  const int hi = lane >> 4, lo = lane & 15;
  const int n = w * 16 + lo;
  v2f bf[64];
#pragma unroll
  for (int kk = 0; kk < 64; ++kk)
    bf[kk] = *(const v2f*)(Amat + (long)n * NDIM + 4 * kk + 2 * hi);
  __shared__ float xbuf[2][16 * NDIM];
  {
    float* xb = &xbuf[0][0];
    for (int idx = threadIdx.x; idx < 2 * 16 * NDIM; idx += 512) xb[idx] = 0.f;
  }
  __syncthreads();
  int cur = 0;
  for (int i = 1; i <= LCHUNK; ++i) {
    const long t = (long)c * LCHUNK + i - 1;
    v8f acc0, acc1;
#pragma unroll
    for (int r = 0; r < 8; ++r) {
      acc0[r] = (hi == 0) ? v[((long)r * SEQ + t) * NDIM + n] : 0.f;
      acc1[r] = 0.f;
    }
#pragma unroll
    for (int kk = 0; kk < 64; kk += 2) {
      const int ks0 = 4 * kk + 2 * hi;
      const v2f a0 = *(const v2f*)&xbuf[cur][lo * NDIM + ks0];
      const v2f a1 = *(const v2f*)&xbuf[cur][lo * NDIM + ks0 + 4];
      acc0 = wmma4(a0, bf[kk], acc0);
      acc1 = wmma4(a1, bf[kk + 1], acc1);
    }
    const int nxt = cur ^ 1;
    if (hi == 0) {
#pragma unroll
      for (int r = 0; r < 8; ++r) {
        const float val = acc0[r] + acc1[r];
        xbuf[nxt][r * NDIM + n] = val;
        states[((long)r * S1 + (long)c * LCHUNK + i) * NDIM + n] = val;
        if (i == LCHUNK) est[((long)c * BSZ + r) * NDIM + n] = val;
      }
    }
    __syncthreads();
    cur = nxt;
  }
}

// ---------------------------------------------------------------------------
// Sequential carry chain over chunks: carry_{c+1} = carry_c * (A^T)^L + e_c.
// B fragments come from row-major A^L (contiguous). Also stores carry_c for
// every chunk and states[:,0,:] = x0.
// ---------------------------------------------------------------------------
__global__ __launch_bounds__(512) void carry_chain(const float* __restrict__ AL,
                                                   const float* __restrict__ est,
                                                   const float* __restrict__ x0,
                                                   float* __restrict__ carr,
                                                   float* __restrict__ states) {
  const int lane = threadIdx.x & 31;
  const int w = threadIdx.x >> 5;
  const int hi = lane >> 4, lo = lane & 15;
  const int n = w * 16 + lo;
  v2f bf[64];
#pragma unroll
  for (int kk = 0; kk < 64; ++kk)
    bf[kk] = *(const v2f*)(AL + (long)n * NDIM + 4 * kk + 2 * hi);
  __shared__ float xbuf[2][16 * NDIM];
  {
    float* xb = &xbuf[0][0];
    for (int idx = threadIdx.x; idx < 2 * 16 * NDIM; idx += 512) xb[idx] = 0.f;
  }
  __syncthreads();
  for (int idx = threadIdx.x; idx < BSZ * NDIM; idx += 512) {
    const float v0 = x0[idx];
    xbuf[0][idx] = v0;                                          // rows 0..7
    states[((long)(idx >> 8) * S1) * NDIM + (idx & 255)] = v0;  // t = 0
  }
  __syncthreads();
  int cur = 0;
  for (int c = 0; c < CCHUNK; ++c) {
    for (int idx = threadIdx.x; idx < BSZ * NDIM; idx += 512)
      carr[(long)c * BSZ * NDIM + idx] = xbuf[cur][idx];
    if (c == CCHUNK - 1) break;
    v8f acc0, acc1;
#pragma unroll
    for (int r = 0; r < 8; ++r) {
      acc0[r] = (hi == 0) ? est[((long)c * BSZ + r) * NDIM + n] : 0.f;
      acc1[r] = 0.f;
    }
#pragma unroll
    for (int kk = 0; kk < 64; kk += 2) {
      const int ks0 = 4 * kk + 2 * hi;
      const v2f a0 = *(const v2f*)&xbuf[cur][lo * NDIM + ks0];
      const v2f a1 = *(const v2f*)&xbuf[cur][lo * NDIM + ks0 + 4];
      acc0 = wmma4(a0, bf[kk], acc0);
      acc1 = wmma4(a1, bf[kk + 1], acc1);
    }
    const int nxt = cur ^ 1;
    if (hi == 0)
#pragma unroll
      for (int r = 0; r < 8; ++r) xbuf[nxt][r * NDIM + n] = acc0[r] + acc1[r];
    __syncthreads();
    cur ^= 1;
  }
}

// ---------------------------------------------------------------------------
// Parallel correction: states[c*L + i] += carry_c * (A^T)^i for all (c, i).
// Carries stacked as [C*8, 256]; B fragments from row-major A^i (contiguous).
// One 16x16 tile per wave, K=256, dual accumulators.
// ---------------------------------------------------------------------------
__global__ __launch_bounds__(256) void correction(const float* __restrict__ carr,
                                                  const float* __restrict__ Apow,
                                                  float* __restrict__ states) {
  const int lane = threadIdx.x & 31;
  const int wv = threadIdx.x >> 5;
  long tile = (long)blockIdx.x * 8 + wv;
  const long total = (long)LCHUNK * (CCHUNK * BSZ / 16) * (NDIM / 16);
  if (tile >= total) return;
  const int nt = (int)(tile % (NDIM / 16));
  tile /= (NDIM / 16);
  const int mt = (int)(tile % (CCHUNK * BSZ / 16));
  const int i = (int)(tile / (CCHUNK * BSZ / 16)) + 1;  // 1..L
  const float* Ai = Apow + (long)(i - 1) * SLOT;        // A^i (row-major)
  const int hi = lane >> 4, lo = lane & 15;
  const int am = mt * 16 + lo;
  const int n = nt * 16 + lo;
  v8f acc0, acc1;
#pragma unroll
  for (int r = 0; r < 8; ++r) {
    const int m = mt * 16 + r + 8 * hi;
    const int cc = m >> 3, b = m & 7;
    acc0[r] = states[((long)b * S1 + (long)cc * LCHUNK + i) * NDIM + n];
    acc1[r] = 0.f;
  }
  const float* aRow = carr + (long)am * NDIM;
  const float* bRow = Ai + (long)n * NDIM;
  for (int k = 0; k < NDIM; k += 8) {
    const int ks0 = k + 2 * hi;
    const v2f a0 = *(const v2f*)(aRow + ks0);
    const v2f a1 = *(const v2f*)(aRow + ks0 + 4);
    const v2f b0 = *(const v2f*)(bRow + ks0);
    const v2f b1 = *(const v2f*)(bRow + ks0 + 4);
    acc0 = wmma4(a0, b0, acc0);
    acc1 = wmma4(a1, b1, acc1);
  }
#pragma unroll
  for (int r = 0; r < 8; ++r) {
    const int m = mt * 16 + r + 8 * hi;
    const int cc = m >> 3, b = m & 7;
    states[((long)b * S1 + (long)cc * LCHUNK + i) * NDIM + n] = acc0[r] + acc1[r];
  }
}

// ---------------------------------------------------------------------------
// Small setup kernels
// ---------------------------------------------------------------------------
__global__ void fill_tri(const float* __restrict__ x11p,
                         const float* __restrict__ x22p,
                         const float* __restrict__ skp, float* __restrict__ X11,
                         float* __restrict__ X22, float* __restrict__ Sk) {
  const int idx = blockIdx.x * blockDim.x + threadIdx.x;
  if (idx >= NDIM * NDIM) return;
  const int i = idx >> 8, j = idx & 255;
  const float lv = (j <= i) ? 1.f : 0.f;
  const int lidx = i * (i + 1) / 2 + ((j <= i) ? j : 0);
  X11[idx] = lv * x11p[lidx];
  X22[idx] = lv * x22p[lidx];
  float s = 0.f;
  if (j > i) s = skp[i * 255 - i * (i - 1) / 2 + (j - i - 1)];
  else if (i > j) s = -skp[j * 255 - j * (j - 1) / 2 + (i - j - 1)];
  Sk[idx] = s;
}

// out = ap*(dap?)*P + aq*(daq?)*Q + ci*(dci?)*I   (P/Q nullable)
__global__ void mat_comb(float* __restrict__ out, const float* __restrict__ P,
                         float ap, const float* dap, const float* __restrict__ Q,
                         float aq, const float* daq, float ci, const float* dci) {
  const int idx = blockIdx.x * blockDim.x + threadIdx.x;
  if (idx >= NDIM * NDIM) return;
  const int r = idx >> 8, cc = idx & 255;
  float v = 0.f;
  if (P) v += ap * (dap ? *dap : 1.f) * P[idx];
  if (Q) v += aq * (daq ? *daq : 1.f) * Q[idx];
  if (r == cc) v += ci * (dci ? *dci : 1.f);
  out[idx] = v;
}

__global__ void neg_sym_eps(const float* __restrict__ R, float* __restrict__ out) {
  const int idx = blockIdx.x * blockDim.x + threadIdx.x;
  if (idx >= NDIM * NDIM) return;
  const int i = idx >> 8, j = idx & 255;
  float v = -0.5f * (R[i * NDIM + j] + R[j * NDIM + i]);
  if (i == j) v += 1e-6f;
  out[idx] = v;
}

__global__ void transposeK(const float* __restrict__ in, float* __restrict__ out) {
  const int idx = blockIdx.x * blockDim.x + threadIdx.x;
  if (idx >= NDIM * NDIM) return;
  const int i = idx >> 8, j = idx & 255;
  out[j * NDIM + i] = in[idx];
}

// Gauss-Jordan inverse (no pivoting; all inputs are well-conditioned here).
__global__ __launch_bounds__(1024) void gauss_jordan(const float* __restrict__ Min,
                                                     float* __restrict__ Mout,
                                                     float* __restrict__ aug) {
  const int N = NDIM, W = 2 * NDIM;
  __shared__ float fac[NDIM];
  for (int idx = threadIdx.x; idx < N * W; idx += 1024) {
    const int r = idx >> 9, cc = idx & (W - 1);
    aug[idx] = (cc < N) ? Min[(long)r * N + cc] : ((cc - N) == r ? 1.f : 0.f);
  }
  __syncthreads();
  for (int p = 0; p < N; ++p) {
    const float inv = 1.f / aug[(long)p * W + p];
    __syncthreads();
    for (int cc = threadIdx.x; cc < W; cc += 1024) aug[(long)p * W + cc] *= inv;
    __syncthreads();
    for (int r = threadIdx.x; r < N; r += 1024) fac[r] = aug[(long)r * W + p];
    __syncthreads();
    for (int idx = threadIdx.x; idx < N * W; idx += 1024) {
      const int r = idx >> 9, cc = idx & (W - 1);
      if (r != p) aug[idx] -= fac[r] * aug[(long)p * W + cc];
    }
    __syncthreads();
  }
  for (int idx = threadIdx.x; idx < N * N; idx += 1024) {
    const int r = idx >> 8, cc = idx & 255;
    Mout[idx] = aug[(long)r * W + NDIM + cc];
  }
}

// Cholesky (lower), right-looking, single block of 256 threads.
__global__ __launch_bounds__(256) void cholesky(const float* __restrict__ Min,
                                                float* __restrict__ L) {
  const int N = NDIM;
  __shared__ float dsh;
  for (int idx = threadIdx.x; idx < N * N; idx += 256) L[idx] = Min[idx];
  __syncthreads();
  const int i = threadIdx.x;
  for (int j = 0; j < N; ++j) {
    if (i == 0) dsh = sqrtf(L[j * N + j]);
    __syncthreads();
    const float d = dsh;
    if (i == j) L[j * N + j] = d;
    if (i > j) L[i * N + j] /= d;
    __syncthreads();
    if (i > j) {
      const float lij = L[i * N + j];
      for (int k = j + 1; k <= i; ++k) L[i * N + k] -= lij * L[k * N + j];
    }
    __syncthreads();
  }
  for (int idx = threadIdx.x; idx < N * N; idx += 256) {
    const int r = idx >> 8, cc = idx & 255;
    if (cc > r) L[idx] = 0.f;
  }
}

// Spectral norm of symmetric PSD Z via power iteration.
__global__ __launch_bounds__(256) void spec_norm(const float* __restrict__ Z,
                                                 float* __restrict__ scal) {
  __shared__ float vsh[NDIM];
  __shared__ float red[NDIM];
  const int i = threadIdx.x;
  vsh[i] = 1.f;
  __syncthreads();
  float nrm = 16.f;
  for (int it = 0; it < 100; ++it) {
    float y = 0.f;
    for (int k = 0; k < NDIM; ++k) y += Z[i * NDIM + k] * vsh[k];
    red[i] = y * y;
    __syncthreads();
    for (int s = 128; s > 0; s >>= 1) {
      if (i < s) red[i] += red[i + s];
      __syncthreads();
    }
    nrm = sqrtf(red[0]);
    __syncthreads();
    vsh[i] = y / nrm;
    __syncthreads();
  }
  if (i == 0) scal[0] = nrm;
}

__global__ void compute_scalars(const float* __restrict__ gammaP,
                                const float* __restrict__ alphaP,
                                float* __restrict__ scal) {
  if (threadIdx.x == 0 && blockIdx.x == 0) {
    const float g = gammaP[0], a = alphaP[0];
    const float sig = 1.f / (1.f + expf(-a));
    const float beta = g * g * sig / scal[0];
    scal[1] = beta;
    scal[2] = sqrtf(beta);
    scal[3] = g * g;
  }
}

// ---------------------------------------------------------------------------
// Host orchestration
// ---------------------------------------------------------------------------
static inline void gg(hipStream_t st, const float* A, long lda, long sA,
                      const float* B, long ldb, long sB, float* C, long ldc,
                      long sC, int M, int N, int K, int tA, int tB, float beta,
                      int batch) {
  const long tiles = (long)(M / 16) * (N / 16);
  dim3 grid((unsigned)((tiles + 7) / 8), 1, (unsigned)batch);
  if (tA) {
    if (tB) gg_kernel<1, 1><<<grid, 256, 0, st>>>(A, lda, sA, B, ldb, sB, C, ldc, sC, M, N, K, beta);
    else gg_kernel<1, 0><<<grid, 256, 0, st>>>(A, lda, sA, B, ldb, sB, C, ldc, sC, M, N, K, beta);
  } else {
    if (tB) gg_kernel<0, 1><<<grid, 256, 0, st>>>(A, lda, sA, B, ldb, sB, C, ldc, sC, M, N, K, beta);
    else gg_kernel<0, 0><<<grid, 256, 0, st>>>(A, lda, sA, B, ldb, sB, C, ldc, sC, M, N, K, beta);
  }
}

extern "C" void kernel_launch(void* const* d_in, const int* in_sizes, int n_in,
                              void* d_out, int out_size, void* d_ws,
                              size_t ws_size, hipStream_t stream) {
  const float* u = (const float*)d_in[0];       // [8, 8192, 256]
  const float* x0 = (const float*)d_in[1];      // [8, 256]
  const float* gammaP = (const float*)d_in[2];  // scalar
  const float* alphaP = (const float*)d_in[3];  // scalar
  const float* x11p = (const float*)d_in[4];
  const float* x22p = (const float*)d_in[5];
  const float* skp = (const float*)d_in[6];
  const float* X21 = (const float*)d_in[7];
  const float* Cp = (const float*)d_in[8];
  const float* Dt = (const float*)d_in[9];

  float* outp = (float*)d_out;                       // [8, 8192, 256]
  float* statesp = outp + (size_t)BSZ * SEQ * NDIM;  // [8, 8193, 256]

  // Workspace layout (floats); total ~93 MB.
  float* ws = (float*)d_ws;
  auto slot = [&](int i) { return ws + (size_t)i * SLOT; };
  // 0 X11, 1 X22, 2 Sk, 3 T1, 4 T2, 5 Qm, 6 Z, 7 H11, 8 H12, 9 V,
  // 10 R, 11 CR, 12 CRH, 13 A, 14 Bm, 15 T3, 16 T4, 17 Dm
  float* scal = ws + (size_t)24 * SLOT;  // [0]=||Z||2 [1]=beta [2]=sqrt(beta) [3]=g^2
  float* aug = scal + 64;                // 256*512 Gauss-Jordan scratch
  float* est = aug + (size_t)NDIM * 2 * NDIM;        // [C][8][256]
  float* carr = est + (size_t)CCHUNK * BSZ * NDIM;   // [C][8][256]
  float* Apow = carr + (size_t)CCHUNK * BSZ * NDIM;  // A^1..A^L (row-major)
  float* vbuf = Apow + (size_t)LCHUNK * SLOT;        // v = u @ B^T, 67 MB

  const dim3 G(256), Blk(256);

  // ---- setup: build A, B, D ----
  fill_tri<<<G, Blk, 0, stream>>>(x11p, x22p, skp, slot(0), slot(1), slot(2));
  mat_comb<<<G, Blk, 0, stream>>>(slot(3), slot(2), 1.f, nullptr, nullptr, 0.f, nullptr, 1.f, nullptr);   // I+Sk
  mat_comb<<<G, Blk, 0, stream>>>(slot(4), slot(2), -1.f, nullptr, nullptr, 0.f, nullptr, 1.f, nullptr);  // I-Sk
  gauss_jordan<<<1, 1024, 0, stream>>>(slot(3), slot(15), aug);  // inv(I+Sk)
  gg(stream, slot(4), NDIM, 0, slot(15), NDIM, 0, slot(5), NDIM, 0, NDIM, NDIM, NDIM, 0, 0, 0.f, 1);  // Qm
  gg(stream, X21, NDIM, 0, X21, NDIM, 0, slot(6), NDIM, 0, NDIM, NDIM, NDIM, 0, 1, 0.f, 1);   // Z = X21 X21^T
  gg(stream, slot(1), NDIM, 0, slot(1), NDIM, 0, slot(6), NDIM, 0, NDIM, NDIM, NDIM, 0, 1, 1.f, 1);  // += X22 X22^T
  gg(stream, Dt, NDIM, 0, Dt, NDIM, 0, slot(6), NDIM, 0, NDIM, NDIM, NDIM, 1, 0, 1.f, 1);     // += Dt^T Dt
  mat_comb<<<G, Blk, 0, stream>>>(slot(6), slot(6), 1.f, nullptr, nullptr, 0.f, nullptr, 1.f, nullptr);  // += I
  spec_norm<<<1, 256, 0, stream>>>(slot(6), scal);
  compute_scalars<<<1, 32, 0, stream>>>(gammaP, alphaP, scal);
  gg(stream, slot(0), NDIM, 0, slot(0), NDIM, 0, slot(7), NDIM, 0, NDIM, NDIM, NDIM, 0, 1, 0.f, 1);  // H11 = X11 X11^T
  gg(stream, Cp, NDIM, 0, Cp, NDIM, 0, slot(7), NDIM, 0, NDIM, NDIM, NDIM, 1, 0, 1.f, 1);            // += C^T C
  mat_comb<<<G, Blk, 0, stream>>>(slot(7), slot(7), 1.f, nullptr, nullptr, 0.f, nullptr, 1.f, scal + 1);  // += beta I
  gg(stream, slot(0), NDIM, 0, X21, NDIM, 0, slot(8), NDIM, 0, NDIM, NDIM, NDIM, 0, 1, 0.f, 1);  // H12 = X11 X21^T
  gg(stream, Cp, NDIM, 0, Dt, NDIM, 0, slot(8), NDIM, 0, NDIM, NDIM, NDIM, 1, 0, 1.f, 1);        // += C^T Dt
  mat_comb<<<G, Blk, 0, stream>>>(slot(8), slot(8), 1.f, scal + 2, nullptr, 0.f, nullptr, 0.f, nullptr);    // *= sqrt(beta)
  mat_comb<<<G, Blk, 0, stream>>>(slot(9), slot(6), 1.f, scal + 1, nullptr, 0.f, nullptr, -1.f, scal + 3);  // V = beta Z - g^2 I
  gauss_jordan<<<1, 1024, 0, stream>>>(slot(9), slot(3), aug);  // inv(V) (V symmetric)
  gg(stream, slot(3), NDIM, 0, slot(8), NDIM, 0, slot(4), NDIM, 0, NDIM, NDIM, NDIM, 0, 1, 0.f, 1);   // S_ = invV H12^T
  gg(stream, slot(8), NDIM, 0, slot(4), NDIM, 0, slot(10), NDIM, 0, NDIM, NDIM, NDIM, 0, 0, 0.f, 1);  // R = H12 S_
  neg_sym_eps<<<G, Blk, 0, stream>>>(slot(10), slot(15));  // negR
  cholesky<<<1, 256, 0, stream>>>(slot(15), slot(11));     // CR
  mat_comb<<<G, Blk, 0, stream>>>(slot(16), slot(15), 1.f, nullptr, slot(7), 1.f, nullptr, 0.f, nullptr);  // negR+H11
  cholesky<<<1, 256, 0, stream>>>(slot(16), slot(12));           // CRH
  gauss_jordan<<<1, 1024, 0, stream>>>(slot(12), slot(3), aug);  // inv(CRH)
  gg(stream, slot(5), NDIM, 0, slot(11), NDIM, 0, slot(4), NDIM, 0, NDIM, NDIM, NDIM, 0, 1, 0.f, 1);   // Qm CR^T
  gg(stream, slot(3), NDIM, 0, slot(4), NDIM, 0, slot(13), NDIM, 0, NDIM, NDIM, NDIM, 1, 0, 0.f, 1);   // A = invCRH^T (Qm CR^T)
  transposeK<<<G, Blk, 0, stream>>>(slot(8), slot(15));           // H12^T
  gauss_jordan<<<1, 1024, 0, stream>>>(slot(15), slot(16), aug);  // inv(H12^T)
  gg(stream, slot(16), NDIM, 0, slot(9), NDIM, 0, slot(15), NDIM, 0, NDIM, NDIM, NDIM, 0, 0, 0.f, 1);  // Xsolve
  gg(stream, slot(13), NDIM, 0, slot(15), NDIM, 0, slot(14), NDIM, 0, NDIM, NDIM, NDIM, 0, 0, 0.f, 1); // Bm = A Xsolve
  mat_comb<<<G, Blk, 0, stream>>>(slot(17), Dt, 1.f, scal + 2, nullptr, 0.f, nullptr, 0.f, nullptr);  // D = sqrt(beta) Dt

  // ---- recurrence: chunked linear scan ----
  gg(stream, u, NDIM, 0, slot(14), NDIM, 0, vbuf, NDIM, 0, BSZ * SEQ, NDIM, NDIM, 0, 1, 0.f, 1);  // v = u B^T
  mat_comb<<<G, Blk, 0, stream>>>(Apow, slot(13), 1.f, nullptr, nullptr, 0.f, nullptr, 0.f, nullptr);  // A^1
  for (int p = 1; p < LCHUNK; p <<= 1) {  // log-depth powers A^2..A^L
    const int cnt = (LCHUNK - p < p) ? (LCHUNK - p) : p;
    gg(stream, Apow, NDIM, SLOT, Apow + (size_t)(p - 1) * SLOT, NDIM, 0,
       Apow + (size_t)p * SLOT, NDIM, SLOT, NDIM, NDIM, NDIM, 0, 0, 0.f, cnt);
  }
  chunk_scan<<<CCHUNK, 512, 0, stream>>>(vbuf, Apow, statesp, est);
  carry_chain<<<1, 512, 0, stream>>>(Apow + (size_t)(LCHUNK - 1) * SLOT, est, x0, carr, statesp);
  correction<<<(LCHUNK * (CCHUNK * BSZ / 16) * (NDIM / 16)) / 8, 256, 0, stream>>>(carr, Apow, statesp);

  // ---- outputs = u D^T + states[:, :-1, :] C^T ----
  gg(stream, u, NDIM, 0, slot(17), NDIM, 0, outp, NDIM, 0, BSZ * SEQ, NDIM, NDIM, 0, 1, 0.f, 1);
  gg(stream, statesp, NDIM, (long)S1 * NDIM, Cp, NDIM, 0, outp, NDIM, (long)SEQ * NDIM,
     SEQ, NDIM, NDIM, 0, 1, 1.f, BSZ);
}